// QuantizedWeight_1726576856662
// MI455X (gfx1250) — compile-verified
//
#include <hip/hip_runtime.h>

// Problem constants (from reference)
#define NOG  4096      // num out groups == out rows (OUT_GROUP==1)
#define NIG  1376      // num in groups
#define IGS  8         // in-group size (floats per decoded cell)
#define IN_F 11008     // in features = NIG * IGS
#define NC   2         // codebooks
#define CS   256       // codebook size
// LDS codebook image: [NC][CS][IGS] floats = 4096 floats = 16 KB
#define CB_FLOATS (NC * CS * IGS)

typedef float v4f __attribute__((ext_vector_type(4)));
typedef int   v2i __attribute__((ext_vector_type(2)));
// GCC-style vector to exactly match the async-builtin's parameter type
typedef int   v4i __attribute__((vector_size(16)));

#if defined(__gfx1250__) && __has_builtin(__builtin_amdgcn_global_load_async_to_lds_b128)
#define USE_ASYNC_LDS 1
// addrspace(1) = global ("__device__"), addrspace(3) = LDS ("__shared__")
typedef __attribute__((address_space(1))) v4i* g4_t;
typedef __attribute__((address_space(3))) v4i* l4_t;
#else
#define USE_ASYNC_LDS 0
#endif

__global__ __launch_bounds__(256) void aqlm_dequant_kernel(
    const int*   __restrict__ codes,      // [NOG, NIG, NC] int32
    const float* __restrict__ codebooks,  // [NC, CS, 1, IGS] fp32
    const float* __restrict__ scales,     // [NOG] fp32
    float*       __restrict__ out)        // [NOG, IN_F] fp32
{
    __shared__ float cb[CB_FLOATS];  // 16 KB, fits easily in 320 KB/WGP

    const int tid = threadIdx.x;

    // ---- Stage codebooks global -> LDS (1024 x 16B chunks, 4 per thread) ----
#if USE_ASYNC_LDS
    {
        #pragma unroll
        for (int i = tid; i < CB_FLOATS / 4; i += 256) {
            __builtin_amdgcn_global_load_async_to_lds_b128(
                (g4_t)(const_cast<float*>(codebooks) + 4 * i),
                (l4_t)(cb + 4 * i),
                /*imm offset*/ 0, /*cpol*/ 0);
        }
#if __has_builtin(__builtin_amdgcn_s_wait_asynccnt)
        __builtin_amdgcn_s_wait_asynccnt(0);
#else
        asm volatile("s_wait_asynccnt 0" ::: "memory");
#endif
    }
#else
    #pragma unroll
    for (int i = tid; i < CB_FLOATS / 4; i += 256) {
        ((v4f*)cb)[i] = ((const v4f*)codebooks)[i];
    }
#endif
    __syncthreads();

    // ---- Decode one output row per block ----
    const int og = blockIdx.x;
    const float s = scales[og];                      // uniform -> s_load
    const v2i*  crow = (const v2i*)(codes + (size_t)og * NIG * NC);
    float*      orow = out + (size_t)og * IN_F;

    for (int ig = tid; ig < NIG; ig += 256) {
        // 8B coalesced, read-once -> non-temporal
        v2i cc = __builtin_nontemporal_load(crow + ig);

        // Gather two 32B codebook entries from LDS (4x ds_load_b128)
        const v4f* e0 = (const v4f*)(cb + (cc.x << 3));              // codebook 0
        const v4f* e1 = (const v4f*)(cb + (CS * IGS) + (cc.y << 3)); // codebook 1
        v4f a0 = e0[0], a1 = e0[1];
        v4f b0 = e1[0], b1 = e1[1];

        v4f r0 = (a0 + b0) * s;
        v4f r1 = (a1 + b1) * s;

        // 32B contiguous per lane, write-once -> non-temporal b128 stores
        v4f* dst = (v4f*)(orow + ig * IGS);
        __builtin_nontemporal_store(r0, dst + 0);
        __builtin_nontemporal_store(r1, dst + 1);
    }
}

extern "C" void kernel_launch(void* const* d_in, const int* in_sizes, int n_in,
                              void* d_out, int out_size, void* d_ws, size_t ws_size,
                              hipStream_t stream) {
    const int*   codes     = (const int*)d_in[0];
    const float* codebooks = (const float*)d_in[1];
    const float* scales    = (const float*)d_in[2];
    float*       out       = (float*)d_out;

    dim3 grid(NOG);    // 4096 blocks, one output row each
    dim3 block(256);   // 8 wave32 waves
    aqlm_dequant_kernel<<<grid, block, 0, stream>>>(codes, codebooks, scales, out);
}